// GNN_36721970380898
// MI455X (gfx1250) — compile-verified
//
#include <hip/hip_runtime.h>
#include <math.h>

typedef float v2f __attribute__((ext_vector_type(2)));
typedef float v8f __attribute__((ext_vector_type(8)));

#define NN    50000
#define NE    400000
#define NVN   50001            // nodes + virtual node
#define ETOT  (NE + NN + NN + NVN)   // 550001 edges total
#define IND   256
#define HID   128
#define LN_EPS 1e-5f
#define NEG_SLOPE 0.2f
#define NEG_ORD 0x007FFFFFu    // f2ord(-inf)

// ---------- helpers ----------
__device__ __forceinline__ unsigned f2ord(float f) {
    unsigned u = __float_as_uint(f);
    return (u & 0x80000000u) ? ~u : (u | 0x80000000u);
}
__device__ __forceinline__ float ord2f(unsigned u) {
    return (u & 0x80000000u) ? __uint_as_float(u & 0x7FFFFFFFu)
                             : __uint_as_float(~u);
}
// edge id -> (src, dst) per reference concatenation:
// [edge_index ; node->vnode ; vnode->node ; self loops incl vnode]
__device__ __forceinline__ void edge_sd(int e, const int* __restrict__ ei0,
                                        const int* __restrict__ ei1,
                                        int& s, int& d) {
    if (e < NE)                { s = ei0[e];        d = ei1[e]; }
    else if (e < NE + NN)      { s = e - NE;        d = NN; }
    else if (e < NE + 2 * NN)  { s = NN;            d = e - (NE + NN); }
    else                       { s = e - (NE + 2 * NN); d = s; }
}

// ---------- vnode mean ----------
__global__ void k_zero(float* p, int n) {
    int i = blockIdx.x * blockDim.x + threadIdx.x;
    if (i < n) p[i] = 0.0f;
}
__global__ void k_colsum(const float* __restrict__ X, float* __restrict__ vsum) {
    int col = threadIdx.x;                 // 256 threads, coalesced across cols
    float acc = 0.0f;
    for (int r = blockIdx.x; r < NN; r += gridDim.x)
        acc += X[(size_t)r * IND + col];
    atomicAdd(&vsum[col], acc);
}
__global__ void k_vnode(const float* __restrict__ vsum, float* __restrict__ xv_last) {
    int t = threadIdx.x;
    xv_last[t] = vsum[t] * (1.0f / (float)NN);
}

// ---------- fp32 WMMA GEMM: C[M,N] = A[M,K] @ B[K,N] (+bias) ----------
// One wave computes a 16x64 tile: one A fragment (16x4 f32) feeds four
// V_WMMA_F32_16X16X4_F32 issues per K-step (full fp32 precision path).
// Requires N % 64 == 0.
__global__ void k_wmma_gemm(const float* __restrict__ A, const float* __restrict__ B,
                            float* __restrict__ C, const float* __restrict__ bias,
                            int nrows, int K, int N, int lda, int ldb, int ldc) {
    int lane = threadIdx.x & 31;
    int wid  = threadIdx.x >> 5;
    int ntN4 = N >> 6;                                 // 64-wide column super-tiles
    int ntM  = (nrows + 15) >> 4;
    int tile = blockIdx.x * (blockDim.x >> 5) + wid;   // wave-uniform
    if (tile >= ntM * ntN4) return;                    // uniform exit, EXEC stays all-1
    int tm = tile / ntN4, tn = tile % ntN4;
    int row0 = tm << 4, col0 = tn << 6;
    int l16  = lane & 15;
    int khalf = (lane >> 4) << 1;                      // K 0/1 (lanes 0-15), 2/3 (16-31)

    int arow = row0 + l16;
    if (arow >= nrows) arow = nrows - 1;               // clamp pad rows (stores masked)
    const float* ap = A + (size_t)arow * lda + khalf;
    const float* bp = B + (size_t)khalf * ldb + (col0 + l16);

    v8f acc0 = {0.f,0.f,0.f,0.f,0.f,0.f,0.f,0.f};
    v8f acc1 = acc0, acc2 = acc0, acc3 = acc0;
    for (int k = 0; k < K; k += 4) {
        v2f a;  a.x = ap[0];       a.y = ap[1];
        v2f b0; b0.x = bp[0];      b0.y = bp[ldb];
        v2f b1; b1.x = bp[16];     b1.y = bp[ldb + 16];
        v2f b2; b2.x = bp[32];     b2.y = bp[ldb + 32];
        v2f b3; b3.x = bp[48];     b3.y = bp[ldb + 48];
        __builtin_prefetch(ap + 16, 0, 1);             // global_prefetch_b8, A stream
        acc0 = __builtin_amdgcn_wmma_f32_16x16x4_f32(false, a, false, b0, (short)0, acc0, false, false);
        acc1 = __builtin_amdgcn_wmma_f32_16x16x4_f32(false, a, false, b1, (short)0, acc1, false, false);
        acc2 = __builtin_amdgcn_wmma_f32_16x16x4_f32(false, a, false, b2, (short)0, acc2, false, false);
        acc3 = __builtin_amdgcn_wmma_f32_16x16x4_f32(false, a, false, b3, (short)0, acc3, false, false);
        ap += 4;
        bp += (size_t)4 * ldb;
    }
    int rbase = row0 + ((lane >> 4) ? 8 : 0);
    v8f accs[4] = {acc0, acc1, acc2, acc3};
    #pragma unroll
    for (int j = 0; j < 4; ++j) {
        int ccol = col0 + (j << 4) + l16;
        float bb = bias ? bias[ccol] : 0.0f;
        #pragma unroll
        for (int r = 0; r < 8; ++r) {
            int rr = rbase + r;
            if (rr < nrows) C[(size_t)rr * ldc + ccol] = accs[j][r] + bb;
        }
    }
}

// ---------- attention logits: AS/AD[n][h] = sum_o H[n,h,o]*a[h,o] ----------
__global__ void k_alphas(const float* __restrict__ H, const float* __restrict__ a_s,
                         const float* __restrict__ a_d,
                         float* __restrict__ AS, float* __restrict__ AD) {
    int n = blockIdx.x, t = threadIdx.x;   // 128 threads
    __shared__ float4 red[128];
    float h0 = H[(size_t)n * 256 + t];
    float h1 = H[(size_t)n * 256 + 128 + t];
    float4 v;
    v.x = h0 * a_s[t];       v.y = h1 * a_s[128 + t];
    v.z = h0 * a_d[t];       v.w = h1 * a_d[128 + t];
    red[t] = v;
    __syncthreads();
    for (int s = 64; s > 0; s >>= 1) {
        if (t < s) {
            red[t].x += red[t + s].x; red[t].y += red[t + s].y;
            red[t].z += red[t + s].z; red[t].w += red[t + s].w;
        }
        __syncthreads();
    }
    if (t == 0) {
        AS[n * 2 + 0] = red[0].x; AS[n * 2 + 1] = red[0].y;
        AD[n * 2 + 0] = red[0].z; AD[n * 2 + 1] = red[0].w;
    }
}

// ---------- per-layer re-init ----------
__global__ void k_layer_init(float* __restrict__ agg, unsigned* __restrict__ mx,
                             float* __restrict__ den) {
    int i = blockIdx.x * blockDim.x + threadIdx.x;
    if (i < NVN * HID) agg[i] = 0.0f;
    if (i < NVN * 2) { mx[i] = NEG_ORD; den[i] = 0.0f; }
}

// ---------- edge pass 1: e = leakyrelu(as[src]+ad[dst]); segment max ----------
__global__ void k_edge_max(const int* __restrict__ ei0, const int* __restrict__ ei1,
                           const float* __restrict__ AS, const float* __restrict__ AD,
                           float* __restrict__ ebuf, unsigned* __restrict__ mx) {
    int tid = blockIdx.x * blockDim.x + threadIdx.x;
    if (tid >= ETOT * 2) return;
    int e = tid >> 1, h = tid & 1;
    int s, d; edge_sd(e, ei0, ei1, s, d);
    float ev = AS[s * 2 + h] + AD[d * 2 + h];
    ev = (ev > 0.0f) ? ev : NEG_SLOPE * ev;
    ebuf[tid] = ev;
    atomicMax(&mx[d * 2 + h], f2ord(ev));
}

// ---------- edge pass 2: ex = exp(e - m[dst]); segment sum ----------
__global__ void k_edge_exp(const int* __restrict__ ei0, const int* __restrict__ ei1,
                           float* __restrict__ ebuf, const unsigned* __restrict__ mx,
                           float* __restrict__ den) {
    int tid = blockIdx.x * blockDim.x + threadIdx.x;
    if (tid >= ETOT * 2) return;
    int e = tid >> 1, h = tid & 1;
    int s, d; edge_sd(e, ei0, ei1, s, d);
    float ex = expf(ebuf[tid] - ord2f(mx[d * 2 + h]));
    ebuf[tid] = ex;
    atomicAdd(&den[d * 2 + h], ex);
}

// ---------- edge pass 3: weighted aggregation, head-mean fused (x0.5) ----------
__global__ void k_edge_agg(const int* __restrict__ ei0, const int* __restrict__ ei1,
                           const float* __restrict__ ebuf, const float* __restrict__ den,
                           const float* __restrict__ H, float* __restrict__ agg) {
    int lane = threadIdx.x & 31;
    int wid  = threadIdx.x >> 5;
    int e = blockIdx.x * (blockDim.x >> 5) + wid;
    if (e >= ETOT) return;
    int s, d; edge_sd(e, ei0, ei1, s, d);
    float a0 = ebuf[e * 2 + 0] / (den[d * 2 + 0] + 1e-16f);
    float a1 = ebuf[e * 2 + 1] / (den[d * 2 + 1] + 1e-16f);
    const float* hs = H + (size_t)s * 256;
    float* ao = agg + (size_t)d * HID;
    #pragma unroll
    for (int j = 0; j < 4; ++j) {
        int o = lane + 32 * j;
        atomicAdd(&ao[o], 0.5f * (a0 * hs[o] + a1 * hs[128 + o]));
    }
}

// ---------- bias + layernorm + exact gelu ----------
__global__ void k_post(const float* __restrict__ agg, const float* __restrict__ b,
                       const float* __restrict__ g, const float* __restrict__ be,
                       float* __restrict__ xn) {
    int n = blockIdx.x, t = threadIdx.x;   // 128 threads
    __shared__ float red[128];
    float v = agg[(size_t)n * HID + t] + b[t];
    red[t] = v; __syncthreads();
    for (int s = 64; s > 0; s >>= 1) { if (t < s) red[t] += red[t + s]; __syncthreads(); }
    float mu = red[0] * (1.0f / HID);
    __syncthreads();
    float dv = v - mu;
    red[t] = dv * dv; __syncthreads();
    for (int s = 64; s > 0; s >>= 1) { if (t < s) red[t] += red[t + s]; __syncthreads(); }
    float var = red[0] * (1.0f / HID);
    float y = dv * rsqrtf(var + LN_EPS) * g[t] + be[t];
    xn[(size_t)n * HID + t] = 0.5f * y * (1.0f + erff(y * 0.70710678118654752f));
}

// ---------- host ----------
extern "C" void kernel_launch(void* const* d_in, const int* in_sizes, int n_in,
                              void* d_out, int out_size, void* d_ws, size_t ws_size,
                              hipStream_t stream) {
    const float* X   = (const float*)d_in[0];
    const int*   ei0 = (const int*)d_in[1];
    const int*   ei1 = ei0 + NE;
    // d_in[2] = edge_weight: unused by the reference
    const float* W[3]  = {(const float*)d_in[3],  (const float*)d_in[9],  (const float*)d_in[15]};
    const float* As[3] = {(const float*)d_in[4],  (const float*)d_in[10], (const float*)d_in[16]};
    const float* Ad[3] = {(const float*)d_in[5],  (const float*)d_in[11], (const float*)d_in[17]};
    const float* Bb[3] = {(const float*)d_in[6],  (const float*)d_in[12], (const float*)d_in[18]};
    const float* Gg[3] = {(const float*)d_in[7],  (const float*)d_in[13], (const float*)d_in[19]};
    const float* Be[3] = {(const float*)d_in[8],  (const float*)d_in[14], (const float*)d_in[20]};
    const float* Wout = (const float*)d_in[21];
    const float* bout = (const float*)d_in[22];
    float* out = (float*)d_out;

    // workspace carve-up (fp32 words)
    float* ws = (float*)d_ws;
    size_t o = 0;
    float*    xv   = ws + o; o += (size_t)NVN * IND;   // layer input (stride 256 L0, 128 later)
    float*    Hbuf = ws + o; o += (size_t)NVN * 256;   // projected features (2 heads x 128)
    float*    agg  = ws + o; o += (size_t)NVN * HID;   // aggregation accumulator
    float*    ebuf = ws + o; o += (size_t)ETOT * 2;    // per-edge-per-head logits / exps
    float*    AS   = ws + o; o += (size_t)NVN * 2;
    float*    AD   = ws + o; o += (size_t)NVN * 2;
    unsigned* MX   = (unsigned*)(ws + o); o += (size_t)NVN * 2;
    float*    DEN  = ws + o; o += (size_t)NVN * 2;
    float*    VSUM = ws + o; o += 256;

    // 1. virtual node = mean over rows; Xv = [X ; vnode]
    k_zero<<<1, 256, 0, stream>>>(VSUM, 256);
    k_colsum<<<128, 256, 0, stream>>>(X, VSUM);
    hipMemcpyAsync(xv, X, (size_t)NN * IND * sizeof(float),
                   hipMemcpyDeviceToDevice, stream);
    k_vnode<<<1, 256, 0, stream>>>(VSUM, xv + (size_t)NN * IND);

    const int eb2 = (ETOT * 2 + 255) / 256;
    const int ebw = (ETOT + 7) / 8;
    const int ib  = (NVN * HID + 255) / 256;

    for (int l = 0; l < 3; ++l) {
        int din = (l == 0) ? IND : HID;
        // H = Xv @ W  (nv x din) @ (din x 256), fp32 WMMA, 16x64 tile per wave
        int tiles = ((NVN + 15) / 16) * (256 / 64);
        k_wmma_gemm<<<(tiles + 7) / 8, 256, 0, stream>>>(
            xv, W[l], Hbuf, nullptr, NVN, din, 256, din, 256, 256);
        k_alphas<<<NVN, 128, 0, stream>>>(Hbuf, As[l], Ad[l], AS, AD);
        k_layer_init<<<ib, 256, 0, stream>>>(agg, MX, DEN);
        k_edge_max<<<eb2, 256, 0, stream>>>(ei0, ei1, AS, AD, ebuf, MX);
        k_edge_exp<<<eb2, 256, 0, stream>>>(ei0, ei1, ebuf, MX, DEN);
        k_edge_agg<<<ebw, 256, 0, stream>>>(ei0, ei1, ebuf, DEN, Hbuf, agg);
        k_post<<<NVN, 128, 0, stream>>>(agg, Bb[l], Gg[l], Be[l], xv); // stride 128 now
    }

    // final projection: nodes (excl. vnode) @ Wout + bout -> d_out
    int ftiles = (NN / 16) * (64 / 64);   // 3125 waves, exact row cover
    k_wmma_gemm<<<(ftiles + 7) / 8, 256, 0, stream>>>(
        xv, Wout, out, bout, NN, HID, 64, HID, 64, 64);
}